// EpisodicSlotMemoryBlock_58617713656462
// MI455X (gfx1250) — compile-verified
//
#include <hip/hip_runtime.h>
#include <stdint.h>

#define KSLOT 32
#define DIM   256
#define XSTR  260   // padded LDS row stride (floats): rows stay 16B-aligned, banks spread
#define XROWS 48    // 32 key rows + q(32) + w(33) + zero pad to 48 for 16x16 tiles

typedef float v2f __attribute__((ext_vector_type(2)));
typedef float v8f __attribute__((ext_vector_type(8)));

static __device__ __forceinline__ float wave_max32(float v) {
#pragma unroll
  for (int m = 16; m >= 1; m >>= 1) v = fmaxf(v, __shfl_xor(v, m, 32));
  return v;
}
static __device__ __forceinline__ float wave_sum32(float v) {
#pragma unroll
  for (int m = 16; m >= 1; m >>= 1) v += __shfl_xor(v, m, 32);
  return v;
}

__global__ __launch_bounds__(256)
void episodic_slot_mem_kernel(
    const float* __restrict__ qv,   const float* __restrict__ wv,
    const float* __restrict__ wstr, const float* __restrict__ keys,
    const float* __restrict__ vals, const float* __restrict__ age,
    const float* __restrict__ strn,
    float* __restrict__ oRead, float* __restrict__ oKeys,
    float* __restrict__ oVals, float* __restrict__ oAge,
    float* __restrict__ oStr) {
  __shared__ float Xs[XROWS * XSTR];
  __shared__ float nsq[34];      // |key_k|^2 (k<32), |q|^2 (32), |w|^2 (33)
  __shared__ float dqs[KSLOT];   // key_k . q
  __shared__ float dws[KSLOT];   // key_k . w
  __shared__ float s_wread[KSLOT];
  __shared__ float s_rate[KSLOT];
  __shared__ float s_dummy[32];  // per-lane sink for branchless tile extraction

  const int tid  = threadIdx.x;
  const int wid  = tid >> 5;
  const int lane = tid & 31;
  const int b    = blockIdx.x;

  const float* keys_b = keys + (size_t)b * (KSLOT * DIM);
  const float* vals_b = vals + (size_t)b * (KSLOT * DIM);
  const float* q_b    = qv   + (size_t)b * DIM;
  const float* w_b    = wv   + (size_t)b * DIM;

  // ---------- Phase 0: async global->LDS staging (CDNA5 ASYNCcnt path) --------
  // keys: 32 rows x 256 floats = 2048 float4 tiles; 8 per thread.
#pragma unroll
  for (int j = 0; j < 8; ++j) {
    int i   = tid + 256 * j;           // float4 tile index
    int row = i >> 6;                  // 64 float4 per row
    int col = (i & 63) << 2;
    uint32_t ldsa = (uint32_t)(uintptr_t)(&Xs[row * XSTR + col]);
    uint32_t voff = (uint32_t)i << 4;  // byte offset into keys_b
    asm volatile("global_load_async_to_lds_b128 %0, %1, %2"
                 :: "v"(ldsa), "v"(voff), "s"(keys_b) : "memory");
  }
  if (wid < 2) {                       // threads 0..63 -> q into row 32
    int i = tid;
    uint32_t ldsa = (uint32_t)(uintptr_t)(&Xs[32 * XSTR + (i << 2)]);
    uint32_t voff = (uint32_t)i << 4;
    asm volatile("global_load_async_to_lds_b128 %0, %1, %2"
                 :: "v"(ldsa), "v"(voff), "s"(q_b) : "memory");
  } else if (wid < 4) {                // threads 64..127 -> w into row 33
    int i = tid - 64;
    uint32_t ldsa = (uint32_t)(uintptr_t)(&Xs[33 * XSTR + (i << 2)]);
    uint32_t voff = (uint32_t)i << 4;
    asm volatile("global_load_async_to_lds_b128 %0, %1, %2"
                 :: "v"(ldsa), "v"(voff), "s"(w_b) : "memory");
  } else {
    // zero rows 34..47 so the (32..47)x(32..47) Gram tile is clean
    for (int idx = tid - 128; idx < 14 * XSTR; idx += 128)
      Xs[34 * XSTR + idx] = 0.0f;
    // warm the vals stream (global_prefetch_b8), one 128B line per iter
    for (int l = tid - 128; l < 256; l += 128)
      __builtin_prefetch(vals_b + l * 32, 0, 3);
  }
  asm volatile("s_wait_asynccnt 0" ::: "memory");
  __syncthreads();

  // ---------- Phase 1: Gram tiles of X (48x256) via V_WMMA_F32_16X16X4_F32 ----
  // Tiles (rowBase,colBase): w0=(0,0) w1=(16,16) -> slot norms (diagonal)
  //                          w2=(0,32) w3=(16,32) -> keys.q (N=0) / keys.w (N=1)
  //                          w4=(32,32) -> |q|^2, |w|^2 (diagonal)
  if (wid < 5) {
    const int rb = (wid == 1 || wid == 3) ? 16 : (wid == 4 ? 32 : 0);
    const int cb = (wid >= 2) ? 32 : (wid == 1 ? 16 : 0);
    const int m16   = lane & 15;
    const int chalf = (lane >> 4) << 1;          // 0 or 2
    const float* Arow = &Xs[(rb + m16) * XSTR];
    const float* Brow = &Xs[(cb + m16) * XSTR];
    v8f c = {};
#pragma unroll 4
    for (int kc = 0; kc < 64; ++kc) {
      const int cc = (kc << 2) + chalf;
      v2f a, bb;
      a.x  = Arow[cc]; a.y  = Arow[cc + 1];
      bb.x = Brow[cc]; bb.y = Brow[cc + 1];
      c = __builtin_amdgcn_wmma_f32_16x16x4_f32(
          /*neg_a=*/false, a, /*neg_b=*/false, bb,
          /*c_mod=*/(short)0, c, /*reuse_a=*/false, /*reuse_b=*/false);
    }
    // C/D layout: VGPR r, lanes 0-15: (M=r, N=lane); lanes 16-31: (M=r+8, N=lane-16)
    // Branchless scatter: cndmask-select destination address, one ds_store per r.
    const int Mbase = (lane < 16) ? 0 : 8;
    const int N = lane & 15;
#pragma unroll
    for (int r = 0; r < 8; ++r) {
      const float v  = c[r];
      const int   gR = rb + Mbase + r;
      const int   gC = cb + N;
      float* dst = &s_dummy[lane];
      if (cb == 32 && rb < 32) {                 // wave-uniform (scalar) branch
        dst = (N == 0) ? &dqs[gR] : dst;
        dst = (N == 1) ? &dws[gR] : dst;
      } else {
        dst = (gR == gC && gR < 34) ? &nsq[gR] : dst;
      }
      *dst = v;
    }
  }
  __syncthreads();

  // ---------- Phase 2: softmaxes, argmax, rates (one wave32 == K slots) -------
  if (wid == 0) {
    const int k = lane;
    const float nk   = sqrtf(nsq[k]);
    const float nq   = sqrtf(nsq[32]);
    const float nw   = sqrtf(nsq[33]);
    const float invk = 1.0f / (nk + 1e-6f);
    const float simr = dqs[k] * invk / (nq + 1e-6f);
    const float simw = dws[k] * invk / (nw + 1e-6f);
    const float ag = age[(size_t)b * KSLOT + k];
    const float st = strn[(size_t)b * KSLOT + k];

    // read softmax
    const float lr = simr + 0.5f * __logf(fminf(fmaxf(st, 0.001f), 1.0f)) - 0.02f * ag;
    const float mr = wave_max32(lr);
    const float er = __expf(lr - mr);
    const float wread = er / wave_sum32(er);

    // write softmax + straight-through hard argmax
    const float lw = simw * 50.0f + 0.05f * log1pf(ag) - 0.5f * st;
    const float mw = wave_max32(lw);
    const float ew = __expf(lw - mw);
    const float soft = ew / wave_sum32(ew);
    const unsigned long long bal = __ballot(lw == mw);
    const int top = __ffsll(bal) - 1;       // first max index (jnp.argmax tie-break)
    const float hard = (k == top) ? 1.0f : 0.0f;
    const float ww = (hard - soft) + soft;  // forward value of straight-through

    const float wsc  = fminf(fmaxf(wstr[b], 0.0f), 1.0f);
    const float rate = ww * wsc * 0.5f;     // WRITE_ALPHA

    s_wread[k] = wread;
    s_rate[k]  = rate;

    oAge[(size_t)b * KSLOT + k] = (ag + 1.0f) * (1.0f - ww);
    const float sdec = st * 0.995f;         // STRENGTH_DECAY
    const float sn   = sdec + ww * wsc * (1.0f - sdec);
    oStr[(size_t)b * KSLOT + k] = fminf(fmaxf(sn, 0.0f), 1.0f);
  }
  __syncthreads();

  // ---------- Phase 3: fused streaming update (vals touched once from HBM) ----
  const int d = tid;                               // 256 threads == DIM
  const float nw    = sqrtf(nsq[33]);
  const float wvd   = Xs[33 * XSTR + d];           // write_vec[d]
  const float wknd  = wvd / (nw + 1e-6f);          // normalized write key
  float acc = 0.0f;
  float* okb = oKeys + (size_t)b * (KSLOT * DIM);
  float* ovb = oVals + (size_t)b * (KSLOT * DIM);
#pragma unroll 4
  for (int k = 0; k < KSLOT; ++k) {
    const float r  = s_rate[k];
    const float wr = s_wread[k];
    const float kvv = Xs[k * XSTR + d];            // keys from LDS (HBM read once)
    __builtin_nontemporal_store((1.0f - r) * kvv + r * wknd, &okb[k * DIM + d]);
    const float v = __builtin_nontemporal_load(&vals_b[k * DIM + d]);
    acc = fmaf(wr, v, acc);
    __builtin_nontemporal_store((1.0f - r) * v + r * wvd, &ovb[k * DIM + d]);
  }
  __builtin_nontemporal_store(acc, &oRead[(size_t)b * DIM + d]);
}

extern "C" void kernel_launch(void* const* d_in, const int* in_sizes, int n_in,
                              void* d_out, int out_size, void* d_ws, size_t ws_size,
                              hipStream_t stream) {
  (void)n_in; (void)out_size; (void)d_ws; (void)ws_size;
  const float* qv   = (const float*)d_in[0];
  const float* wv   = (const float*)d_in[1];
  const float* wstr = (const float*)d_in[2];
  const float* keys = (const float*)d_in[3];
  const float* vals = (const float*)d_in[4];
  const float* age  = (const float*)d_in[5];
  const float* strn = (const float*)d_in[6];

  const int B = in_sizes[2];                 // write_strength has B elements
  const size_t BD  = (size_t)B * DIM;
  const size_t BKD = (size_t)B * KSLOT * DIM;
  const size_t BK  = (size_t)B * KSLOT;

  float* out   = (float*)d_out;
  float* oRead = out;
  float* oKeys = out + BD;
  float* oVals = out + BD + BKD;
  float* oAge  = out + BD + 2 * BKD;
  float* oStr  = out + BD + 2 * BKD + BK;

  episodic_slot_mem_kernel<<<dim3(B), dim3(256), 0, stream>>>(
      qv, wv, wstr, keys, vals, age, strn, oRead, oKeys, oVals, oAge, oStr);
}